// GaussPSF_44263932953325
// MI455X (gfx1250) — compile-verified
//
#include <hip/hip_runtime.h>
#include <stdint.h>

#define HH    512
#define WW    512
#define BATCH 4
#define CHN   3
#define RAD   5
#define KS    11
#define TX    32
#define TYO   16                 /* output rows per block (2 per thread-row) */
#define HX   (TX + 2*RAD)        /* 42 */
#define HY   (TYO + 2*RAD)       /* 26 */
#define NHALO (HX*HY)            /* 1092 */
#define EPSF  1e-5f
#define PLANE ((size_t)HH * WW)  /* 262144 */

// Raw LDS byte offset of a shared-memory object (generic -> addrspace(3) cast).
__device__ __forceinline__ uint32_t lds_addr_of(const void* p) {
  return (uint32_t)(uintptr_t)(const __attribute__((address_space(3))) char*)p;
}

__global__ __launch_bounds__(256, 1)
void gauss_psf_sep_kernel(const float* __restrict__ img,
                          const float* __restrict__ psf,
                          float* __restrict__ out)
{
  __shared__ float4 sImg[NHALO];         // [HY][HX], .x/.y/.z = channels 0..2
  __shared__ float  sE[HY * 6 * HX];     // [HY][a=0..5][HX] separable exp planes

  const int tid = threadIdx.x;
  const int b   = blockIdx.z;
  const int bx  = blockIdx.x * TX;
  const int by  = blockIdx.y * TYO;
  const int gx0 = bx - RAD;
  const int gy0 = by - RAD;

  const uint32_t ldsImg = lds_addr_of(&sImg[0]);

  // ---------- Stage 1: halo fill ----------
  // Image tile: per-lane async global->LDS B32 copies (ASYNCcnt path).
  // Addresses are clamped into the image so no garbage/NaN bits are read;
  // out-of-bounds taps are killed by E==0 (the reference's mask mv).
  for (int i = tid; i < NHALO; i += 256) {
    const int hy = i / HX;
    const int hx = i - hy * HX;
    const int gy = gy0 + hy;
    const int gx = gx0 + hx;
    const bool ib = ((unsigned)gy < (unsigned)HH) & ((unsigned)gx < (unsigned)WW);
    const int cy = min(max(gy, 0), HH - 1);
    const int cx = min(max(gx, 0), WW - 1);

    const uint32_t voff = (uint32_t)((cy * WW + cx) * 4);
    const uint32_t lb   = ldsImg + (uint32_t)i * 16u;
    #pragma unroll
    for (int c = 0; c < CHN; ++c) {
      const float* plane = img + ((size_t)(b * CHN + c)) * PLANE;
      asm volatile("global_load_async_to_lds_b32 %0, %1, %2"
                   :: "v"(lb + (uint32_t)(c * 4)), "v"(voff), "s"(plane)
                   : "memory");
    }

    // psf -> separable factors E[a] = exp(-a^2 / (2*sigma^2 + eps)), a = 0..5.
    const float sg  = psf[((size_t)b * HH + cy) * WW + cx];
    const float inv = 1.0f / (2.0f * sg * sg + EPSF);
    float* ep = &sE[(hy * 6) * HX + hx];
    #pragma unroll
    for (int a = 0; a < 6; ++a) {
      const float e = (a == 0) ? 1.0f : __expf(-(float)(a * a) * inv);
      ep[a * HX] = ib ? e : 0.0f;
    }
  }

  asm volatile("s_wait_asynccnt 0" ::: "memory");  // own-wave async loads done
  __syncthreads();                                  // all waves' LDS writes visible

  // ---------- Stage 2: 11x11 gather, 2 adjacent output rows per thread ----------
  // Shared neighbor rows: img float4 and ex factor identical for both outputs.
  // Rolled row loop bounds the live set (no scratch spills); inner 11-tap loop
  // stays unrolled so ex offsets are compile-time and ey loads vectorize.
  const int tx = tid & (TX - 1);
  const int ty = tid >> 5;        // 0..7
  const int r0 = 2 * ty;          // first halo row of this thread's 12-row window

  float a0 = 0.f, a1 = 0.f, a2 = 0.f, w0s = 0.f;   // output row oy
  float b0 = 0.f, b1 = 0.f, b2 = 0.f, w1s = 0.f;   // output row oy+1

  { // peeled row j = 0: contributes to output 0 only, ady0 = 5
    const float*  exb  = &sE[(r0 * 6) * HX + tx];
    const float*  e0   = exb + 5 * HX;
    const float4* irow = &sImg[r0 * HX + tx];
    #pragma unroll
    for (int dx = 0; dx < KS; ++dx) {
      const int adx = (dx < RAD) ? (RAD - dx) : (dx - RAD);
      const float w = e0[dx] * exb[adx * HX + dx];
      const float4 iv = irow[dx];
      w0s += w;
      a0 = fmaf(iv.x, w, a0);
      a1 = fmaf(iv.y, w, a1);
      a2 = fmaf(iv.z, w, a2);
    }
  }

  #pragma unroll 1
  for (int j = 1; j <= 10; ++j) {                  // steady state: both outputs
    const int ny = r0 + j;
    const int d5 = j - 5, d6 = j - 6;
    const int ady0 = (d5 < 0) ? -d5 : d5;
    const int ady1 = (d6 < 0) ? -d6 : d6;
    const float*  exb  = &sE[(ny * 6) * HX + tx];
    const float*  e0   = exb + ady0 * HX;
    const float*  e1   = exb + ady1 * HX;
    const float4* irow = &sImg[ny * HX + tx];
    #pragma unroll
    for (int dx = 0; dx < KS; ++dx) {
      const int adx = (dx < RAD) ? (RAD - dx) : (dx - RAD);
      const float ex  = exb[adx * HX + dx];
      const float4 iv = irow[dx];
      const float w0 = e0[dx] * ex;
      const float w1 = e1[dx] * ex;
      w0s += w0;
      a0 = fmaf(iv.x, w0, a0);
      a1 = fmaf(iv.y, w0, a1);
      a2 = fmaf(iv.z, w0, a2);
      w1s += w1;
      b0 = fmaf(iv.x, w1, b0);
      b1 = fmaf(iv.y, w1, b1);
      b2 = fmaf(iv.z, w1, b2);
    }
  }

  { // peeled row j = 11: contributes to output 1 only, ady1 = 5
    const int ny = r0 + 11;
    const float*  exb  = &sE[(ny * 6) * HX + tx];
    const float*  e1   = exb + 5 * HX;
    const float4* irow = &sImg[ny * HX + tx];
    #pragma unroll
    for (int dx = 0; dx < KS; ++dx) {
      const int adx = (dx < RAD) ? (RAD - dx) : (dx - RAD);
      const float w = e1[dx] * exb[adx * HX + dx];
      const float4 iv = irow[dx];
      w1s += w;
      b0 = fmaf(iv.x, w, b0);
      b1 = fmaf(iv.y, w, b1);
      b2 = fmaf(iv.z, w, b2);
    }
  }

  const float rn0 = 1.0f / w0s;
  const float rn1 = 1.0f / w1s;
  const int gy = by + 2 * ty;
  const int gx = bx + tx;
  const size_t o = (((size_t)b * CHN) * HH + gy) * WW + gx;
  out[o]                 = a0 * rn0;
  out[o + WW]            = b0 * rn1;
  out[o + PLANE]         = a1 * rn0;
  out[o + PLANE + WW]    = b1 * rn1;
  out[o + 2*PLANE]       = a2 * rn0;
  out[o + 2*PLANE + WW]  = b2 * rn1;
}

extern "C" void kernel_launch(void* const* d_in, const int* in_sizes, int n_in,
                              void* d_out, int out_size, void* d_ws, size_t ws_size,
                              hipStream_t stream) {
  (void)in_sizes; (void)n_in; (void)out_size; (void)d_ws; (void)ws_size;
  const float* img = (const float*)d_in[0];   // (4,3,512,512) f32
  const float* psf = (const float*)d_in[1];   // (4,512,512)   f32
  float* out = (float*)d_out;                 // (4,3,512,512) f32

  dim3 grid(WW / TX, HH / TYO, BATCH);        // (16, 32, 4)
  gauss_psf_sep_kernel<<<grid, 256, 0, stream>>>(img, psf, out);
}